// UEDDIEMoE_13511967113391
// MI455X (gfx1250) — compile-verified
//
#include <hip/hip_runtime.h>
#include <math.h>

typedef __attribute__((ext_vector_type(2))) float v2f;
typedef __attribute__((ext_vector_type(8))) float v8f;
typedef int vec4i __attribute__((vector_size(16)));   // matches async builtin param type

#define DM      512   // D_MODEL
#define NATOM   512   // N (atoms per batch)
#define NBATCH  128   // B
#define NCOL    64    // 8 gate + 32 (8x4 elem) + 24 (8x3 chg)
#define NWAVES  16
#define BLOCK   (NWAVES * 32)
#define KC      32    // K-chunk staged per double-buffer slot
#define SPAD    34    // padded row stride (floats), stage view  -> conflict-free A reads
#define YPAD    65    // padded row stride (floats), epilogue view

#define UBUF_FLOATS (2 * NWAVES * 32 * SPAD)   // 34816 floats = 139264 B (>= Ys view 133120 B)

// ---- async global -> LDS (gfx1250) ----
__device__ __forceinline__ void async_ld_b128(const float* g, float* l) {
#if __has_builtin(__builtin_amdgcn_global_load_async_to_lds_b128)
    __builtin_amdgcn_global_load_async_to_lds_b128(
        (__attribute__((address_space(1))) vec4i*)(g),
        (__attribute__((address_space(3))) vec4i*)(l), 0, 0);
#else
    unsigned loff = (unsigned)(unsigned long long)
        (__attribute__((address_space(3))) void*)(l);
    asm volatile("global_load_async_to_lds_b128 %0, %1, off"
                 :: "v"(loff), "v"(g) : "memory");
#endif
}

__device__ __forceinline__ void wait_async_le8() {
#if __has_builtin(__builtin_amdgcn_s_wait_asynccnt)
    __builtin_amdgcn_s_wait_asynccnt(8);
#else
    asm volatile("s_wait_asynccnt 0x8" ::: "memory");
#endif
    asm volatile("" ::: "memory");
}

__device__ __forceinline__ void wait_async_0() {
#if __has_builtin(__builtin_amdgcn_s_wait_asynccnt)
    __builtin_amdgcn_s_wait_asynccnt(0);
#else
    asm volatile("s_wait_asynccnt 0x0" ::: "memory");
#endif
    asm volatile("" ::: "memory");
}

// packed weight column j of the fused [DM x 64] weight matrix
__device__ __forceinline__ float wcol(const float* __restrict__ Wg,
                                      const float* __restrict__ We,
                                      const float* __restrict__ Wc,
                                      int k, int j) {
    if (j < 8)  return Wg[k * 8 + j];
    if (j < 40) return We[(j - 8)  * DM + k];
    return             Wc[(j - 40) * DM + k];
}

// One workgroup per batch b. 16 waves x 32 atoms = 512 atoms = exactly one batch.
__global__ __launch_bounds__(BLOCK)
void ueddie_moe_kernel(const float* __restrict__ X,
                       const int*   __restrict__ E,
                       const int*   __restrict__ C,
                       const float* __restrict__ Wg,   // [DM, 8]
                       const float* __restrict__ bg,   // [8]
                       const float* __restrict__ We,   // [8,4,DM]
                       const float* __restrict__ be,   // [8,4]
                       const float* __restrict__ Wc,   // [8,3,DM]
                       const float* __restrict__ bc,   // [8,3]
                       float* __restrict__ out)        // [B]
{
    // K-pair interleaved weights: Wtp[k/2][j] = (W[k][j], W[k+1][j])
    // -> each WMMA B-fragment is ONE naturally aligned ds_load_b64.
    __shared__ v2f   Wtp[DM / 2][NCOL];                // 128 KB
    __shared__ __align__(16) float UBUF[UBUF_FLOATS];  // 136 KB: X stage <-> Y scratch (aliased)
    __shared__ float Gsum[8];
    __shared__ float EOsum[8];

    const int tid  = threadIdx.x;
    const int lane = tid & 31;
    const int wave = tid >> 5;
    const int b    = blockIdx.x;

    // ---- Stage 1: pack weight columns into LDS (K-pair interleaved) ----
    for (int idx = tid; idx < (DM / 2) * NCOL; idx += BLOCK) {
        const int kp = idx >> 6;        // K-pair index
        const int j  = idx & 63;        // fused column
        v2f w;
        w.x = wcol(Wg, We, Wc, 2 * kp,     j);
        w.y = wcol(Wg, We, Wc, 2 * kp + 1, j);
        Wtp[kp][j] = w;
    }
    if (tid < 8) { Gsum[tid] = 0.f; EOsum[tid] = 0.f; }

    // ---- Stage 2: double-buffered async staging of X + WMMA GEMM ----
    const int atom0 = wave * 32;                 // this wave's 32 atoms
    const float* Xb = X + (size_t)b * NATOM * DM;

    // async-fill mapping: instruction i, lane l moves 16B:
    //   row = i*4 + l/8, col = (l%8)*4   (4 rows x 128 contiguous bytes each)
    const int srow = lane >> 3;
    const int scol = (lane & 7) << 2;

    auto issue_chunk = [&](int buf, int kc) {
#pragma unroll
        for (int i = 0; i < 8; ++i) {
            const int row = i * 4 + srow;
            const float* g = Xb + (size_t)(atom0 + row) * DM + kc + scol;
            float* l = &UBUF[(size_t)((buf * NWAVES + wave) * 32 + row) * SPAD + scol];
            async_ld_b128(g, l);
        }
    };

    issue_chunk(0, 0);          // prime buffer 0 (stage region disjoint from Wtp)
    __syncthreads();            // Wtp ready

    // fp32 WMMA 16x16x4 fragment mapping (wave32):
    //  A: lane l holds M = l%16, K = k0 + (l<16 ? 0 : 2) + {0,1}
    //  B: lane l holds N = nt*16 + l%16, same K pairs
    const int mrow  = lane & 15;
    const int koff  = (lane >> 4) << 1;  // 0 or 2  (A-fragment float offset)
    const int khalf = lane >> 4;         // 0 or 1  (B-fragment K-pair offset)

    v8f zero = {0.f, 0.f, 0.f, 0.f, 0.f, 0.f, 0.f, 0.f};
    v8f acc0[4], acc1[4];
#pragma unroll
    for (int nt = 0; nt < 4; ++nt) { acc0[nt] = zero; acc1[nt] = zero; }

#pragma unroll 1
    for (int c = 0; c < DM / KC; ++c) {
        const int buf = c & 1;
        if (c + 1 < DM / KC) {
            issue_chunk(buf ^ 1, (c + 1) * KC);  // prefetch next chunk
            wait_async_le8();                    // in-order completion => chunk c resident
        } else {
            wait_async_0();
        }

        const float* sb = &UBUF[(size_t)(buf * NWAVES + wave) * 32 * SPAD];
        const int kb = c * KC;
#pragma unroll
        for (int k0 = 0; k0 < KC; k0 += 4) {
            v2f a0 = *(const v2f*)(sb + mrow * SPAD + k0 + koff);
            v2f a1 = *(const v2f*)(sb + (16 + mrow) * SPAD + k0 + koff);
            const int kp = ((kb + k0) >> 1) + khalf;   // K-pair row in Wtp
#pragma unroll
            for (int nt = 0; nt < 4; ++nt) {
                v2f bf = Wtp[kp][nt * 16 + mrow];      // single ds_load_b64
                acc0[nt] = __builtin_amdgcn_wmma_f32_16x16x4_f32(
                    false, a0, false, bf, (short)0, acc0[nt], false, false);
                acc1[nt] = __builtin_amdgcn_wmma_f32_16x16x4_f32(
                    false, a1, false, bf, (short)0, acc1[nt], false, false);
            }
        }
    }

    // ---- Stage 3: re-purpose UBUF as Y scratch (views overlap across waves) ----
    __syncthreads();

    float* ys = &UBUF[(size_t)wave * 32 * YPAD];
    const int mhalf = (lane >> 4) << 3;  // 0 or 8
#pragma unroll
    for (int nt = 0; nt < 4; ++nt) {
#pragma unroll
        for (int r = 0; r < 8; ++r) {
            ys[(r + mhalf)      * YPAD + nt * 16 + mrow] = acc0[nt][r];
            ys[(16 + r + mhalf) * YPAD + nt * 16 + mrow] = acc1[nt][r];
        }
    }

    // ---- Stage 4: per-atom epilogue (one lane per atom; wave-local LDS dep) ----
    {
        const int n = atom0 + lane;                 // atom index within batch
        const float* row = ys + (size_t)lane * YPAD;
        const int e = E[b * NATOM + n];             // element label 0..3
        const int c = C[b * NATOM + n];             // charge label  0..2

        float gl[8], mx = -1e30f;
#pragma unroll
        for (int x = 0; x < 8; ++x) { gl[x] = row[x] + bg[x]; mx = fmaxf(mx, gl[x]); }
        float se = 0.f;
#pragma unroll
        for (int x = 0; x < 8; ++x) { gl[x] = expf(gl[x] - mx); se += gl[x]; }
        const float inv = 1.f / se;

#pragma unroll
        for (int x = 0; x < 8; ++x) {
            const float g  = gl[x] * inv;
            const float pe = row[8  + x * 4 + e] + be[x * 4 + e];
            const float pc = row[40 + x * 3 + c] + bc[x * 3 + c];
            atomicAdd(&Gsum[x],  g);
            atomicAdd(&EOsum[x], pe * fabsf(pc));
        }
    }
    __syncthreads();

    // ---- Stage 5: out[b] = -sum_x Gsum[x] * EOsum[x] ----
    if (tid == 0) {
        float s = 0.f;
#pragma unroll
        for (int x = 0; x < 8; ++x) s += Gsum[x] * EOsum[x];
        out[b] = -s;
    }
}

extern "C" void kernel_launch(void* const* d_in, const int* in_sizes, int n_in,
                              void* d_out, int out_size, void* d_ws, size_t ws_size,
                              hipStream_t stream) {
    const float* X  = (const float*)d_in[0];
    const int*   E  = (const int*)  d_in[1];
    const int*   C  = (const int*)  d_in[2];
    const float* Wg = (const float*)d_in[3];
    const float* bg = (const float*)d_in[4];
    const float* We = (const float*)d_in[5];
    const float* be = (const float*)d_in[6];
    const float* Wc = (const float*)d_in[7];
    const float* bc = (const float*)d_in[8];
    float* out = (float*)d_out;

    hipLaunchKernelGGL(ueddie_moe_kernel, dim3(NBATCH), dim3(BLOCK), 0, stream,
                       X, E, C, Wg, bg, We, be, Wc, bc, out);
}